// AugmentPipe_39213051413152
// MI455X (gfx1250) — compile-verified
//
#include <hip/hip_runtime.h>
#include <hip/hip_bf16.h>

// MI455X / gfx1250 augmentation gather.
// Pure data movement: ~201MB traffic -> ~8.6us floor @ 23.3 TB/s.
// All 4 reference stages composed into one per-pixel index remap; transpose
// samples staged through LDS via gfx1250 async global->LDS (ASYNCcnt path).

#define IMG_H 256
#define IMG_W 256
#define IMG_C 3
#define TILE  32
#define PITCH (TILE * IMG_C + 3)   // 99 floats: odd stride mod 64 banks -> conflict-free

#if defined(__HIP_DEVICE_COMPILE__) && __has_builtin(__builtin_amdgcn_global_load_async_to_lds_b32)
#define HAVE_ASYNC_LDS 1
#else
#define HAVE_ASYNC_LDS 0
#endif

#if HAVE_ASYNC_LDS
// Probe round told us param 0 is a non-const global ('__device__') int*.
typedef __attribute__((address_space(3))) int lds_int_t;
typedef __attribute__((address_space(1))) int glob_int_t;
__device__ __forceinline__ lds_int_t* to_lds(void* p) {
    // generic shared ptr low 32 bits == LDS byte offset on gfx1250
    return (lds_int_t*)(unsigned int)(unsigned long long)p;
}
__device__ __forceinline__ glob_int_t* to_glob(const void* p) {
    return (glob_int_t*)(unsigned long long)p;
}
#endif

__device__ __forceinline__ int reflect_idx(int v) {
    // v in [-32, 287] here; python-mod into [0, 509], then reflect
    int m = v % 510;
    if (m < 0) m += 510;
    return 255 - abs(255 - m);
}

__global__ __launch_bounds__(256) void augment_kernel(
    const float* __restrict__ img,
    const int*   __restrict__ xflip_w, const float* __restrict__ xflip_gate,
    const int*   __restrict__ yflip_w, const float* __restrict__ yflip_gate,
    const int*   __restrict__ rot_w,   const float* __restrict__ rot_gate,
    const float* __restrict__ trans_w, const float* __restrict__ trans_gate,
    float* __restrict__ out, int N)
{
    __shared__ float lds[TILE * PITCH];

    const int n   = blockIdx.z;
    const int ty0 = blockIdx.y * TILE;
    const int tx0 = blockIdx.x * TILE;

    // ---- per-sample uniform params (scalar loads) ----
    const int xf = (xflip_gate[n] < 1.0f) ? xflip_w[n] : 0;
    const int yf = (yflip_gate[n] < 1.0f) ? yflip_w[n] : 0;
    const int rv = (rot_gate[n]   < 1.0f) ? rot_w[n]   : 0;
    const int rfW   = (rv == 1) | (rv == 2);
    const int rfH   = (rv == 2) | (rv == 3);
    const int tpose = rv & 1;                      // rv==1 || rv==3

    const float g   = trans_gate[n];
    const float tw0 = (g < 1.0f) ? (trans_w[n]     * 2.0f - 1.0f) : 0.0f;
    const float tw1 = (g < 1.0f) ? (trans_w[N + n] * 2.0f - 1.0f) : 0.0f;
    const int tx = (int)rintf(tw0 * 32.0f);        // W * TRANSLATE_INT_MAX = 32
    const int ty = (int)rintf(tw1 * 32.0f);

    const int fY = rfH ^ (yf == 1);
    const int fX = rfW ^ (xf == 1);

    const float* __restrict__ src = img + (size_t)n * (IMG_H * IMG_W * IMG_C);
    float*       __restrict__ dst = out + (size_t)n * (IMG_H * IMG_W * IMG_C);

    if (!tpose) {
        // Direct separable gather: lane-fastest along xo -> coalesced in and out.
        #pragma unroll
        for (int j = 0; j < 4; ++j) {
            const int l  = (int)threadIdx.x + j * 256;
            const int yo = ty0 + (l >> 5);
            const int xo = tx0 + (l & 31);
            const int p  = reflect_idx(yo + ty);   // yi[yo]
            const int q  = reflect_idx(xo - tx);   // xi[xo]
            const int yin = fY ? 255 - p : p;
            const int xin = fX ? 255 - q : q;
            const float* sp = src + ((size_t)yin * IMG_W + xin) * IMG_C;
            float*       dp = dst + ((size_t)yo  * IMG_W + xo ) * IMG_C;
            __builtin_nontemporal_store(sp[0], dp + 0);
            __builtin_nontemporal_store(sp[1], dp + 1);
            __builtin_nontemporal_store(sp[2], dp + 2);
        }
    } else {
        // Transposed sample: out[yo][xo] = in[y_in(xi[xo])][x_in(yi[yo])]
        // Phase 1: lane-fastest along yo (the input-contiguous direction),
        // gather straight into LDS with async global->LDS (no VGPR staging).
        #pragma unroll
        for (int j = 0; j < 4; ++j) {
            const int l = (int)threadIdx.x + j * 256;
            const int a = l >> 5;                  // xo_local
            const int b = l & 31;                  // yo_local (lane-fastest)
            const int p = reflect_idx((tx0 + a) - tx);   // xi[xo]
            const int q = reflect_idx((ty0 + b) + ty);   // yi[yo]
            const int yin = fY ? 255 - p : p;
            const int xin = fX ? 255 - q : q;
            const float* sp = src + ((size_t)yin * IMG_W + xin) * IMG_C;
            float*       lp = &lds[a * PITCH + b * IMG_C];
#if HAVE_ASYNC_LDS
            __builtin_amdgcn_global_load_async_to_lds_b32(to_glob(sp), to_lds(lp), 0, 0);
            __builtin_amdgcn_global_load_async_to_lds_b32(to_glob(sp), to_lds(lp), 4, 0);
            __builtin_amdgcn_global_load_async_to_lds_b32(to_glob(sp), to_lds(lp), 8, 0);
#else
            lp[0] = sp[0];
            lp[1] = sp[1];
            lp[2] = sp[2];
#endif
        }
#if HAVE_ASYNC_LDS
#if __has_builtin(__builtin_amdgcn_s_wait_asynccnt)
        __builtin_amdgcn_s_wait_asynccnt(0);
#else
        asm volatile("s_wait_asynccnt 0" ::: "memory");
#endif
#endif
        __syncthreads();
        // Phase 2: lane-fastest along xo -> coalesced NT stores.
        // LDS read stride per lane = PITCH(=99) floats, odd mod 64 banks -> no conflicts.
        #pragma unroll
        for (int j = 0; j < 4; ++j) {
            const int l = (int)threadIdx.x + j * 256;
            const int b = l >> 5;                  // yo_local
            const int a = l & 31;                  // xo_local (lane-fastest)
            const float* lp = &lds[a * PITCH + b * IMG_C];
            float*       dp = dst + ((size_t)(ty0 + b) * IMG_W + (tx0 + a)) * IMG_C;
            __builtin_nontemporal_store(lp[0], dp + 0);
            __builtin_nontemporal_store(lp[1], dp + 1);
            __builtin_nontemporal_store(lp[2], dp + 2);
        }
    }
}

extern "C" void kernel_launch(void* const* d_in, const int* in_sizes, int n_in,
                              void* d_out, int out_size, void* d_ws, size_t ws_size,
                              hipStream_t stream) {
    const float* images     = (const float*)d_in[0];
    const int*   xflip_w    = (const int*)  d_in[1];
    const float* xflip_gate = (const float*)d_in[2];
    const int*   yflip_w    = (const int*)  d_in[3];
    const float* yflip_gate = (const float*)d_in[4];
    const int*   rot_w      = (const int*)  d_in[5];
    const float* rot_gate   = (const float*)d_in[6];
    const float* trans_w    = (const float*)d_in[7];
    const float* trans_gate = (const float*)d_in[8];
    float* out = (float*)d_out;

    const int N = in_sizes[1];                     // 128 samples
    dim3 grid(IMG_W / TILE, IMG_H / TILE, N);      // 8 x 8 x 128 blocks
    dim3 block(256);                               // 8 wave32 waves
    augment_kernel<<<grid, block, 0, stream>>>(
        images, xflip_w, xflip_gate, yflip_w, yflip_gate,
        rot_w, rot_gate, trans_w, trans_gate, out, N);
}